// ControlWhile_35983236006544
// MI455X (gfx1250) — compile-verified
//
#include <hip/hip_runtime.h>
#include <math.h>

// ---------------------------------------------------------------------------
// Problem constants (from the reference)
// ---------------------------------------------------------------------------
#define BATCH   8
#define CIN     3
#define COUTC   16
#define HW      147456                 // 384*384
#define NTILES  73728                  // BATCH*HW / 16 pixels-per-tile
#define TPB     8                      // tiles (waves) per 256-thread block
#define NBLK    (NTILES / TPB)         // 9216 blocks, no remainder
#define NELEM   (BATCH * COUTC * HW)   // 18,874,368 floats = 75.5 MB (fits L2)
#define MAXIT   12                     // upper bound on while-loop iterations
#define QSHIFT  20                     // fixed-point scale for deterministic mean
#define THRESH_ULL ((3ULL * (unsigned long long)NELEM) << QSHIFT)

typedef __attribute__((ext_vector_type(2))) float v2f;
typedef __attribute__((ext_vector_type(8))) float v8f;

// D = A(16x4 f32) * B(4x16 f32) + C(16x16 f32), one WMMA per K-chunk of 4.
__device__ __forceinline__ v8f wmma_k4(v2f a, v2f b, v8f c) {
    return __builtin_amdgcn_wmma_f32_16x16x4_f32(
        /*neg_a=*/false, a, /*neg_b=*/false, b,
        /*c_mod=*/(short)0, c, /*reuse_a=*/false, /*reuse_b=*/false);
}

__device__ __forceinline__ unsigned long long qabs(float x) {
    return (unsigned long long)(fabsf(x) * (float)(1u << QSHIFT));
}

// ---------------------------------------------------------------------------
// init: zero the per-iteration accumulators and flags in workspace
// ---------------------------------------------------------------------------
__global__ void init_ws_kernel(unsigned long long* acc, int* flags) {
    int t = threadIdx.x;
    if (t < MAXIT) { acc[t] = 0ULL; flags[t] = 0; }
}

// ---------------------------------------------------------------------------
// pre-conv: v = W_pre[16x3] * x[3ch] + b_pre   (single WMMA, K padded 3->4)
// Tile = 16 consecutive pixels in one batch plane; one wave per tile.
// ---------------------------------------------------------------------------
__global__ __launch_bounds__(256) void pre_kernel(
        const float* __restrict__ x, float* __restrict__ v,
        const float* __restrict__ wP, const float* __restrict__ bP) {
    __shared__ float ldsW[64];                       // 16x4, col 3 zero-padded
    int t = threadIdx.x;
    if (t < 64) {
        int o = t >> 2, c = t & 3;
        ldsW[t] = (c < CIN) ? wP[o * CIN + c] : 0.0f;
    }
    __syncthreads();

    int wave = t >> 5, lane = t & 31;
    int tile = blockIdx.x * TPB + wave;
    int bidx = tile / (HW / 16);
    int hw0  = (tile % (HW / 16)) * 16;
    const float* xb = x + bidx * CIN   * HW + hw0;
    float*       vb = v + bidx * COUTC * HW + hw0;

    int half = lane >> 4;                            // 0 or 1
    int n    = lane & 15;                            // pixel within tile / A-row

    v8f acc;
    #pragma unroll
    for (int r = 0; r < 8; ++r) acc[r] = bP[r + half * 8];

    int kk = half * 2;                               // K-pair: {0,1} or {2,3}
    v2f a; a.x = ldsW[n * 4 + kk]; a.y = ldsW[n * 4 + kk + 1];
    v2f b; b.x = xb[kk * HW + n];
    b.y = (kk + 1 < CIN) ? xb[(kk + 1) * HW + n] : 0.0f;   // pad row 3 (A=0 there)
    acc = wmma_k4(a, b, acc);

    #pragma unroll
    for (int r = 0; r < 8; ++r) vb[(r + half * 8) * HW + n] = acc[r];
}

// ---------------------------------------------------------------------------
// reduce: deterministic fixed-point sum of |v| (integer atomics = associative)
// ---------------------------------------------------------------------------
__global__ __launch_bounds__(256) void reduce_kernel(
        const float* __restrict__ v, unsigned long long* __restrict__ acc,
        const int* __restrict__ flags, int iter) {
    if (iter > 0 && flags[iter - 1]) return;         // already converged: skip
    __shared__ unsigned long long sdata[256];
    const float4* v4 = (const float4*)v;
    int n4 = NELEM / 4;
    unsigned long long s = 0;
    for (int i = blockIdx.x * blockDim.x + threadIdx.x; i < n4;
         i += gridDim.x * blockDim.x) {
        float4 f = v4[i];
        s += qabs(f.x) + qabs(f.y) + qabs(f.z) + qabs(f.w);
    }
    sdata[threadIdx.x] = s;
    __syncthreads();
    for (int off = 128; off > 0; off >>= 1) {
        if (threadIdx.x < off) sdata[threadIdx.x] += sdata[threadIdx.x + off];
        __syncthreads();
    }
    if (threadIdx.x == 0) atomicAdd(&acc[iter], sdata[0]);
}

__global__ void finalize_kernel(const unsigned long long* acc, int* flags, int iter) {
    // while (mean < 3) body  =>  done when sum_fixed >= 3*N*2^20
    int done = (iter > 0 && flags[iter - 1]) ? 1 : (acc[iter] >= THRESH_ULL ? 1 : 0);
    flags[iter] = done;
}

// ---------------------------------------------------------------------------
// body (in-place): v = 10 * (W_loop * tanh(W_shared * v + b_shared) + b_loop)
// Two chained 16x16 f32 GEMMs per 16-pixel tile; D->B relayout via per-wave LDS.
// ---------------------------------------------------------------------------
__global__ __launch_bounds__(256) void body_kernel(
        float* __restrict__ v,
        const float* __restrict__ wS, const float* __restrict__ bS,
        const float* __restrict__ wL, const float* __restrict__ bL,
        const int* __restrict__ flags, int iter) {
    if (flags[iter]) return;                          // uniform early-exit

    __shared__ float ldsWS[256];
    __shared__ float ldsWL[256];
    __shared__ float stage[TPB][256];                 // per-wave 16ch x 16px
    int t = threadIdx.x;
    ldsWS[t] = wS[t];
    ldsWL[t] = wL[t];
    __syncthreads();

    int wave = t >> 5, lane = t & 31;
    int tile = blockIdx.x * TPB + wave;
    int bidx = tile / (HW / 16);
    int hw0  = (tile % (HW / 16)) * 16;
    float* base = v + bidx * COUTC * HW + hw0;

    int half = lane >> 4;
    int n    = lane & 15;

    // ---- GEMM 1: W_shared * v + b_shared -------------------------------
    v8f c1;
    #pragma unroll
    for (int r = 0; r < 8; ++r) c1[r] = bS[r + half * 8];
    #pragma unroll
    for (int k0 = 0; k0 < COUTC; k0 += 4) {
        int kk = k0 + half * 2;
        v2f a; a.x = ldsWS[n * 16 + kk]; a.y = ldsWS[n * 16 + kk + 1];
        v2f b; b.x = base[kk * HW + n];  b.y = base[(kk + 1) * HW + n];
        c1 = wmma_k4(a, b, c1);
    }

    // ---- tanh, then stage D-layout -> (ch,px) in per-wave LDS ----------
    #pragma unroll
    for (int r = 0; r < 8; ++r)
        stage[wave][(r + half * 8) * 16 + n] = tanhf(c1[r]);
    // wave-local RAW across lanes: wait for our DS stores, order reads after
    asm volatile("s_wait_dscnt 0x0" ::: "memory");

    // ---- GEMM 2: W_loop * tanh(.) + b_loop, then *10, store in place ---
    v8f c2;
    #pragma unroll
    for (int r = 0; r < 8; ++r) c2[r] = bL[r + half * 8];
    #pragma unroll
    for (int k0 = 0; k0 < COUTC; k0 += 4) {
        int kk = k0 + half * 2;
        v2f a; a.x = ldsWL[n * 16 + kk];      a.y = ldsWL[n * 16 + kk + 1];
        v2f b; b.x = stage[wave][kk * 16 + n]; b.y = stage[wave][(kk + 1) * 16 + n];
        c2 = wmma_k4(a, b, c2);
    }
    #pragma unroll
    for (int r = 0; r < 8; ++r)
        base[(r + half * 8) * HW + n] = c2[r] * 10.0f;
}

// ---------------------------------------------------------------------------
// final conv (in-place): out = W_shared * v + b_shared
// ---------------------------------------------------------------------------
__global__ __launch_bounds__(256) void final_kernel(
        float* __restrict__ v,
        const float* __restrict__ wS, const float* __restrict__ bS) {
    __shared__ float ldsW[256];
    int t = threadIdx.x;
    ldsW[t] = wS[t];
    __syncthreads();

    int wave = t >> 5, lane = t & 31;
    int tile = blockIdx.x * TPB + wave;
    int bidx = tile / (HW / 16);
    int hw0  = (tile % (HW / 16)) * 16;
    float* base = v + bidx * COUTC * HW + hw0;

    int half = lane >> 4;
    int n    = lane & 15;

    v8f acc;
    #pragma unroll
    for (int r = 0; r < 8; ++r) acc[r] = bS[r + half * 8];
    #pragma unroll
    for (int k0 = 0; k0 < COUTC; k0 += 4) {
        int kk = k0 + half * 2;
        v2f a; a.x = ldsW[n * 16 + kk]; a.y = ldsW[n * 16 + kk + 1];
        v2f b; b.x = base[kk * HW + n]; b.y = base[(kk + 1) * HW + n];
        acc = wmma_k4(a, b, acc);
    }
    #pragma unroll
    for (int r = 0; r < 8; ++r)
        base[(r + half * 8) * HW + n] = acc[r];
}

// ---------------------------------------------------------------------------
// launch: pre -> MAXIT x {reduce, finalize, cond-body} -> final   (all in-place
// on d_out, v stays resident in the 192MB L2; d_ws holds only acc/flags)
// ---------------------------------------------------------------------------
extern "C" void kernel_launch(void* const* d_in, const int* in_sizes, int n_in,
                              void* d_out, int out_size, void* d_ws, size_t ws_size,
                              hipStream_t stream) {
    const float* x        = (const float*)d_in[0];
    const float* w_pre    = (const float*)d_in[1];
    const float* b_pre    = (const float*)d_in[2];
    const float* w_loop   = (const float*)d_in[3];
    const float* b_loop   = (const float*)d_in[4];
    const float* w_shared = (const float*)d_in[5];
    const float* b_shared = (const float*)d_in[6];
    float* out = (float*)d_out;

    unsigned long long* acc = (unsigned long long*)d_ws;
    int* flags = (int*)((char*)d_ws + MAXIT * sizeof(unsigned long long));

    init_ws_kernel<<<1, 32, 0, stream>>>(acc, flags);
    pre_kernel<<<NBLK, 256, 0, stream>>>(x, out, w_pre, b_pre);

    for (int i = 0; i < MAXIT; ++i) {
        reduce_kernel<<<1024, 256, 0, stream>>>(out, acc, flags, i);
        finalize_kernel<<<1, 1, 0, stream>>>(acc, flags, i);
        body_kernel<<<NBLK, 256, 0, stream>>>(out, w_shared, b_shared,
                                              w_loop, b_loop, flags, i);
    }

    final_kernel<<<NBLK, 256, 0, stream>>>(out, w_shared, b_shared);
}